// MultiDirectionalSpatialScanner_82351702934177
// MI455X (gfx1250) — compile-verified
//
#include <hip/hip_runtime.h>
#include <hip/hip_bf16.h>

// ---------------- problem constants ----------------
#define DN   1024     // model dim
#define NN   576      // tokens (24*24)
#define BB   4        // batch
#define HH   8        // heads
#define HDD  128      // head dim
#define KDIR 8        // scan directions
#define KNN  4608     // KDIR * NN
#define BNN  2304     // BB * NN
#define BKND 18432    // BB * KDIR * NN

typedef float v2f __attribute__((ext_vector_type(2)));
typedef float v8f __attribute__((ext_vector_type(8)));

// ---------------- GEMM tiling ----------------
#define TILE_M 64
#define TILE_N 64
#define KC     32
#define LDT    36   // padded LDS stride (fragment reads conflict-free, 16B-aligned rows)

// A staged as Als[m][k]; B staged n-major as Bls[n][k] so both WMMA fragments are
// contiguous 8-byte LDS loads:  a[j]=A[m][kk+j],  b[j]=B[kk+j][n]=Bls[n][kk+j].
// 256 threads = 8 waves; wave w owns a 16x32 sub-tile. Double-buffered K chunks.

// ===================== dirs GEMM: kv = gather(vision) @ dir_W[k] + dir_b[k] =====================
__global__ __launch_bounds__(256)
void dirs_gemm_kernel(const float* __restrict__ vision, const int* __restrict__ scan_idx,
                      const float* __restrict__ dirW, const float* __restrict__ dirb,
                      float* __restrict__ kv)
{
    __shared__ float Als[2][TILE_M][LDT];
    __shared__ float Bls[2][TILE_N][LDT];

    const int b = blockIdx.z >> 3;
    const int k = blockIdx.z & 7;
    const float* A  = vision + (size_t)b * NN * DN;
    const float* Bm = dirW + (size_t)k * DN * DN;
    const int*   idx = scan_idx + k * NN;
    float* C = kv + ((size_t)b * KNN + (size_t)k * NN) * DN;

    const int m0 = blockIdx.x * TILE_M;
    const int n0 = blockIdx.y * TILE_N;
    const int tid  = threadIdx.x;
    const int lane = tid & 31;
    const int wave = tid >> 5;
    const int lm   = lane & 15;
    const int lg   = lane >> 4;
    const int wm   = (wave & 3) * 16;
    const int wn   = (wave >> 2) * 32;

    const int arow  = tid >> 3;            // 0..31
    const int ac4   = (tid & 7) * 4;       // 0..28
    const int grow0 = idx[m0 + arow];
    const int grow1 = idx[m0 + arow + 32];
    const int brow  = tid >> 4;            // 0..15 (k index in B source)
    const int bc4   = (tid & 15) * 4;      // 0..60 (n index in B source)

    v8f acc0 = {}; v8f acc1 = {};

    // prologue: stage chunk 0
    float4 a0r = *(const float4*)(A + (size_t)grow0 * DN + ac4);
    float4 a1r = *(const float4*)(A + (size_t)grow1 * DN + ac4);
    float4 w0r = *(const float4*)(Bm + (size_t)brow * DN + n0 + bc4);
    float4 w1r = *(const float4*)(Bm + (size_t)(brow + 16) * DN + n0 + bc4);
    *(float4*)&Als[0][arow][ac4]      = a0r;
    *(float4*)&Als[0][arow + 32][ac4] = a1r;
    Bls[0][bc4 + 0][brow] = w0r.x; Bls[0][bc4 + 1][brow] = w0r.y;
    Bls[0][bc4 + 2][brow] = w0r.z; Bls[0][bc4 + 3][brow] = w0r.w;
    Bls[0][bc4 + 0][brow + 16] = w1r.x; Bls[0][bc4 + 1][brow + 16] = w1r.y;
    Bls[0][bc4 + 2][brow + 16] = w1r.z; Bls[0][bc4 + 3][brow + 16] = w1r.w;
    __syncthreads();

    const int nIter = DN / KC;
    for (int it = 0; it < nIter; ++it) {
        const int nxt = it + 1;
        if (nxt < nIter) {
            const int k0 = nxt * KC;
            a0r = *(const float4*)(A + (size_t)grow0 * DN + k0 + ac4);
            a1r = *(const float4*)(A + (size_t)grow1 * DN + k0 + ac4);
            w0r = *(const float4*)(Bm + (size_t)(k0 + brow) * DN + n0 + bc4);
            w1r = *(const float4*)(Bm + (size_t)(k0 + brow + 16) * DN + n0 + bc4);
        }
        const int cb = it & 1;
        #pragma unroll
        for (int s = 0; s < KC / 4; ++s) {
            const int kk = s * 4 + lg * 2;
            v2f af = *(const v2f*)&Als[cb][wm + lm][kk];
            v2f b0 = *(const v2f*)&Bls[cb][wn + lm][kk];
            v2f b1 = *(const v2f*)&Bls[cb][wn + 16 + lm][kk];
            acc0 = __builtin_amdgcn_wmma_f32_16x16x4_f32(false, af, false, b0, (short)0, acc0, false, false);
            acc1 = __builtin_amdgcn_wmma_f32_16x16x4_f32(false, af, false, b1, (short)0, acc1, false, false);
        }
        if (nxt < nIter) {
            const int nb = nxt & 1;
            *(float4*)&Als[nb][arow][ac4]      = a0r;
            *(float4*)&Als[nb][arow + 32][ac4] = a1r;
            Bls[nb][bc4 + 0][brow] = w0r.x; Bls[nb][bc4 + 1][brow] = w0r.y;
            Bls[nb][bc4 + 2][brow] = w0r.z; Bls[nb][bc4 + 3][brow] = w0r.w;
            Bls[nb][bc4 + 0][brow + 16] = w1r.x; Bls[nb][bc4 + 1][brow + 16] = w1r.y;
            Bls[nb][bc4 + 2][brow + 16] = w1r.z; Bls[nb][bc4 + 3][brow + 16] = w1r.w;
            __syncthreads();
        }
    }

    const float bv0 = dirb[(size_t)k * DN + n0 + wn + lm];
    const float bv1 = dirb[(size_t)k * DN + n0 + wn + 16 + lm];
    #pragma unroll
    for (int r = 0; r < 8; ++r) {
        const int row = m0 + wm + r + 8 * lg;
        C[(size_t)row * DN + n0 + wn + lm]      = acc0[r] + bv0;
        C[(size_t)row * DN + n0 + wn + 16 + lm] = acc1[r] + bv1;
    }
}

// ===================== generic projection: C = A @ W^T + bias =====================
// W is (E, DN) row-major, so the n-major LDS staging is a direct float4 copy.
__global__ __launch_bounds__(256)
void proj_gemm_kernel(const float* __restrict__ A, const float* __restrict__ W,
                      const float* __restrict__ bias, float* __restrict__ C)
{
    __shared__ float Als[2][TILE_M][LDT];
    __shared__ float Bls[2][TILE_N][LDT];

    const int m0 = blockIdx.x * TILE_M;
    const int n0 = blockIdx.y * TILE_N;
    const int tid  = threadIdx.x;
    const int lane = tid & 31;
    const int wave = tid >> 5;
    const int lm   = lane & 15;
    const int lg   = lane >> 4;
    const int wm   = (wave & 3) * 16;
    const int wn   = (wave >> 2) * 32;

    const int arow = tid >> 3;           // 0..31
    const int ac4  = (tid & 7) * 4;      // 0..28

    v8f acc0 = {}; v8f acc1 = {};

    float4 a0r = *(const float4*)(A + (size_t)(m0 + arow) * DN + ac4);
    float4 a1r = *(const float4*)(A + (size_t)(m0 + arow + 32) * DN + ac4);
    float4 b0r = *(const float4*)(W + (size_t)(n0 + arow) * DN + ac4);
    float4 b1r = *(const float4*)(W + (size_t)(n0 + arow + 32) * DN + ac4);
    *(float4*)&Als[0][arow][ac4]      = a0r;
    *(float4*)&Als[0][arow + 32][ac4] = a1r;
    *(float4*)&Bls[0][arow][ac4]      = b0r;
    *(float4*)&Bls[0][arow + 32][ac4] = b1r;
    __syncthreads();

    const int nIter = DN / KC;
    for (int it = 0; it < nIter; ++it) {
        const int nxt = it + 1;
        if (nxt < nIter) {
            const int k0 = nxt * KC;
            a0r = *(const float4*)(A + (size_t)(m0 + arow) * DN + k0 + ac4);
            a1r = *(const float4*)(A + (size_t)(m0 + arow + 32) * DN + k0 + ac4);
            b0r = *(const float4*)(W + (size_t)(n0 + arow) * DN + k0 + ac4);
            b1r = *(const float4*)(W + (size_t)(n0 + arow + 32) * DN + k0 + ac4);
        }
        const int cb = it & 1;
        #pragma unroll
        for (int s = 0; s < KC / 4; ++s) {
            const int kk = s * 4 + lg * 2;
            v2f af = *(const v2f*)&Als[cb][wm + lm][kk];
            v2f b0 = *(const v2f*)&Bls[cb][wn + lm][kk];
            v2f b1 = *(const v2f*)&Bls[cb][wn + 16 + lm][kk];
            acc0 = __builtin_amdgcn_wmma_f32_16x16x4_f32(false, af, false, b0, (short)0, acc0, false, false);
            acc1 = __builtin_amdgcn_wmma_f32_16x16x4_f32(false, af, false, b1, (short)0, acc1, false, false);
        }
        if (nxt < nIter) {
            const int nb = nxt & 1;
            *(float4*)&Als[nb][arow][ac4]      = a0r;
            *(float4*)&Als[nb][arow + 32][ac4] = a1r;
            *(float4*)&Bls[nb][arow][ac4]      = b0r;
            *(float4*)&Bls[nb][arow + 32][ac4] = b1r;
            __syncthreads();
        }
    }

    const float bv0 = bias[n0 + wn + lm];
    const float bv1 = bias[n0 + wn + 16 + lm];
    #pragma unroll
    for (int r = 0; r < 8; ++r) {
        const int row = m0 + wm + r + 8 * lg;
        C[(size_t)row * DN + n0 + wn + lm]      = acc0[r] + bv0;
        C[(size_t)row * DN + n0 + wn + 16 + lm] = acc1[r] + bv1;
    }
}

// ===================== flash attention =====================
#define QT  32       // query rows per block
#define CKC 32       // keys per chunk
#define LDQ 132      // padded stride for 128-wide LDS rows
#define LDP 36       // padded stride for P / Vt rows

__global__ __launch_bounds__(128)
void attn_kernel(const float* __restrict__ Qb, const float* __restrict__ Kb,
                 const float* __restrict__ Vb, float* __restrict__ Ob)
{
    __shared__ float Qs[QT][LDQ];
    __shared__ float Ks[CKC][LDQ];
    __shared__ float Vt[HDD][LDP];     // V transposed: Vt[col][key]
    __shared__ float Ps[QT][LDP];
    __shared__ float m_run[QT], l_run[QT], m_new_s[QT], scale_s[QT];
    __shared__ float smax[2][QT], ssum[2][QT];

    const int bh = blockIdx.y;
    const int b  = bh >> 3;
    const int h  = bh & 7;
    const int q0 = blockIdx.x * QT;

    const float* Q  = Qb + (size_t)b * NN  * DN + (size_t)h * HDD;
    const float* Kp = Kb + (size_t)b * KNN * DN + (size_t)h * HDD;
    const float* Vp = Vb + (size_t)b * KNN * DN + (size_t)h * HDD;

    const int tid  = threadIdx.x;
    const int lane = tid & 31;
    const int wave = tid >> 5;
    const int lm   = lane & 15;
    const int lg   = lane >> 4;
    const int qh   = wave & 1;   // which 16 query rows
    const int kh   = wave >> 1;  // which 16 keys of chunk / which 64-col half of O

    for (int i = tid; i < QT * 32; i += 128) {
        const int row = i >> 5, c4 = (i & 31) * 4;
        *(float4*)&Qs[row][c4] = *(const float4*)(Q + (size_t)(q0 + row) * DN + c4);
    }
    if (tid < QT) { m_run[tid] = -3.402823466e38f; l_run[tid] = 0.f; }
    v8f o_acc[4] = {{}, {}, {}, {}};
    __syncthreads();

    const float sm_scale = 0.08838834764831845f;  // 1/sqrt(128)

    for (int c0 = 0; c0 < KNN; c0 += CKC) {
        for (int i = tid; i < CKC * 32; i += 128) {
            const int row = i >> 5, c4 = (i & 31) * 4;
            *(float4*)&Ks[row][c4] = *(const float4*)(Kp + (size_t)(c0 + row) * DN + c4);
            float4 v4 = *(const float4*)(Vp + (size_t)(c0 + row) * DN + c4);
            Vt[c4 + 0][row] = v4.x; Vt[c4 + 1][row] = v4.y;
            Vt[c4 + 2][row] = v4.z; Vt[c4 + 3][row] = v4.w;
        }
        __syncthreads();

        // ---- S tile: 16 q-rows (qh) x 16 keys (kh), K=HD=128 ----
        v8f s_acc = {};
        #pragma unroll
        for (int st = 0; st < HDD / 4; ++st) {
            const int kk = st * 4 + lg * 2;
            v2f af = *(const v2f*)&Qs[qh * 16 + lm][kk];
            v2f bf = *(const v2f*)&Ks[kh * 16 + lm][kk];  // B[d][n] = K[n][d]
            s_acc = __builtin_amdgcn_wmma_f32_16x16x4_f32(false, af, false, bf, (short)0, s_acc, false, false);
        }
        #pragma unroll
        for (int r = 0; r < 8; ++r) s_acc[r] *= sm_scale;

        // per-row chunk max (across the 16 lanes of each half)
        #pragma unroll
        for (int r = 0; r < 8; ++r) {
            float v = s_acc[r];
            v = fmaxf(v, __shfl_xor(v, 1, 32));
            v = fmaxf(v, __shfl_xor(v, 2, 32));
            v = fmaxf(v, __shfl_xor(v, 4, 32));
            v = fmaxf(v, __shfl_xor(v, 8, 32));
            if (lm == 0) smax[kh][qh * 16 + r + 8 * lg] = v;
        }
        __syncthreads();

        if (tid < QT) {
            const float mo = m_run[tid];
            const float mc = fmaxf(smax[0][tid], smax[1][tid]);
            const float mn = fmaxf(mo, mc);
            m_new_s[tid] = mn;
            scale_s[tid] = __expf(mo - mn);
            m_run[tid]   = mn;
        }
        __syncthreads();

        // ---- P = exp(S - m_new) -> LDS, partial row sums ----
        #pragma unroll
        for (int r = 0; r < 8; ++r) {
            const int row = qh * 16 + r + 8 * lg;
            const float p = __expf(s_acc[r] - m_new_s[row]);
            Ps[row][kh * 16 + lm] = p;
            float v = p;
            v += __shfl_xor(v, 1, 32);
            v += __shfl_xor(v, 2, 32);
            v += __shfl_xor(v, 4, 32);
            v += __shfl_xor(v, 8, 32);
            if (lm == 0) ssum[kh][row] = v;
        }
        // rescale O accumulators
        #pragma unroll
        for (int t = 0; t < 4; ++t)
            #pragma unroll
            for (int r = 0; r < 8; ++r)
                o_acc[t][r] *= scale_s[qh * 16 + r + 8 * lg];
        __syncthreads();

        if (tid < QT)
            l_run[tid] = l_run[tid] * scale_s[tid] + ssum[0][tid] + ssum[1][tid];

        // ---- O += P @ V : 16 rows (qh) x 64 cols (kh), K = 32 keys ----
        #pragma unroll
        for (int st = 0; st < CKC / 4; ++st) {
            const int kk = st * 4 + lg * 2;
            v2f af = *(const v2f*)&Ps[qh * 16 + lm][kk];
            #pragma unroll
            for (int t = 0; t < 4; ++t) {
                const int col = kh * 64 + t * 16 + lm;
                v2f bf = *(const v2f*)&Vt[col][kk];   // B[key][col] = Vt[col][key]
                o_acc[t] = __builtin_amdgcn_wmma_f32_16x16x4_f32(false, af, false, bf, (short)0, o_acc[t], false, false);
            }
        }
        __syncthreads();
    }

    #pragma unroll
    for (int t = 0; t < 4; ++t) {
        #pragma unroll
        for (int r = 0; r < 8; ++r) {
            const int row = qh * 16 + r + 8 * lg;
            const float inv = 1.0f / l_run[row];
            float* orow = Ob + ((size_t)b * NN + q0 + row) * DN + (size_t)h * HDD;
            orow[kh * 64 + t * 16 + lm] = o_acc[t][r] * inv;
        }
    }
}

// ===================== LayerNorm + residual =====================
__global__ __launch_bounds__(256)
void ln_residual_kernel(const float* __restrict__ y, const float* __restrict__ g,
                        const float* __restrict__ bta, const float* __restrict__ vision,
                        float* __restrict__ out)
{
    __shared__ float red[16];
    const int row = blockIdx.x;
    const int tid = threadIdx.x;
    const float* yr = y + (size_t)row * DN;

    float s = 0.f, s2 = 0.f;
    for (int c = tid; c < DN; c += 256) { const float v = yr[c]; s += v; s2 += v * v; }
    for (int off = 16; off > 0; off >>= 1) {
        s  += __shfl_xor(s,  off, 32);
        s2 += __shfl_xor(s2, off, 32);
    }
    if ((tid & 31) == 0) { red[tid >> 5] = s; red[8 + (tid >> 5)] = s2; }
    __syncthreads();
    if (tid == 0) {
        float ts = 0.f, ts2 = 0.f;
        #pragma unroll
        for (int i = 0; i < 8; ++i) { ts += red[i]; ts2 += red[8 + i]; }
        const float mu  = ts * (1.0f / DN);
        const float var = ts2 * (1.0f / DN) - mu * mu;
        red[0] = mu;
        red[1] = rsqrtf(var + 1e-5f);
    }
    __syncthreads();
    const float mu = red[0], rstd = red[1];
    const float* vr = vision + (size_t)row * DN;
    float* orow = out + (size_t)row * DN;
    for (int c = tid; c < DN; c += 256)
        orow[c] = (yr[c] - mu) * rstd * g[c] + bta[c] + vr[c];
}

// ===================== host =====================
extern "C" void kernel_launch(void* const* d_in, const int* in_sizes, int n_in,
                              void* d_out, int out_size, void* d_ws, size_t ws_size,
                              hipStream_t stream) {
    const float* vision = (const float*)d_in[0];
    const float* dirW   = (const float*)d_in[1];
    const float* dirb   = (const float*)d_in[2];
    const float* in_w   = (const float*)d_in[3];
    const float* in_b   = (const float*)d_in[4];
    const float* outw   = (const float*)d_in[5];
    const float* outb   = (const float*)d_in[6];
    const float* finw   = (const float*)d_in[7];
    const float* finb   = (const float*)d_in[8];
    const float* ln_g   = (const float*)d_in[9];
    const float* ln_b   = (const float*)d_in[10];
    const int*   scan   = (const int*)d_in[11];
    float* out = (float*)d_out;

    float* ws    = (float*)d_ws;
    float* kvbuf = ws;                                   // 18432*1024
    float* qbuf  = kvbuf + (size_t)BKND * DN;            // 2304*1024
    float* kbuf  = qbuf  + (size_t)BNN * DN;             // 18432*1024
    float* vbuf  = kbuf  + (size_t)BKND * DN;            // 18432*1024
    // aliases into kvbuf once it has been consumed (stream-ordered):
    float* obuf  = kvbuf;
    float* fbuf  = kvbuf + (size_t)BNN * DN;
    float* ybuf  = kvbuf + (size_t)2 * BNN * DN;

    dim3 blk(256);

    // 1. kv = gather(vision) @ dir_W + dir_b
    dirs_gemm_kernel<<<dim3(NN / TILE_M, DN / TILE_N, BB * KDIR), blk, 0, stream>>>(
        vision, scan, dirW, dirb, kvbuf);

    // 2. q = vision @ wq^T + bq
    proj_gemm_kernel<<<dim3(BNN / TILE_M, DN / TILE_N), blk, 0, stream>>>(
        vision, in_w, in_b, qbuf);

    // 3. k = kv @ wk^T + bk ; v = kv @ wv^T + bv
    proj_gemm_kernel<<<dim3(BKND / TILE_M, DN / TILE_N), blk, 0, stream>>>(
        kvbuf, in_w + (size_t)DN * DN, in_b + DN, kbuf);
    proj_gemm_kernel<<<dim3(BKND / TILE_M, DN / TILE_N), blk, 0, stream>>>(
        kvbuf, in_w + (size_t)2 * DN * DN, in_b + 2 * DN, vbuf);

    // 4. flash attention -> o
    attn_kernel<<<dim3(NN / QT, BB * HH), dim3(128), 0, stream>>>(qbuf, kbuf, vbuf, obuf);

    // 5. fused = o @ out_proj_w^T + b ; y = fused @ fin_w^T + b
    proj_gemm_kernel<<<dim3(BNN / TILE_M, DN / TILE_N), blk, 0, stream>>>(
        obuf, outw, outb, fbuf);
    proj_gemm_kernel<<<dim3(BNN / TILE_M, DN / TILE_N), blk, 0, stream>>>(
        fbuf, finw, finb, ybuf);

    // 6. LayerNorm + residual
    ln_residual_kernel<<<dim3(BNN), blk, 0, stream>>>(ybuf, ln_g, ln_b, vision, out);
}